// Hungarian_match_67199058313515
// MI455X (gfx1250) — compile-verified
//
#include <hip/hip_runtime.h>

// Problem constants from the reference
#define NB 16            // batch
#define NS 50            // src/tgt sequence length
#define NP 36864         // 192*192 flattened mask

#define KSPLIT 16                 // waves per output tile (K-split factor)
#define KSLICE (NP / KSPLIT)      // 2304 K-elements per wave

typedef float v2f __attribute__((ext_vector_type(2)));
typedef float v8f __attribute__((ext_vector_type(8)));

// ---------------------------------------------------------------------------
// Kernel 1: row sum-of-squares  ws[0..800) = s2, ws[800..1600) = t2
// One block per row; 128-thread strided float4 reduction (bandwidth pass,
// per-batch data stays hot in the 192MB L2 for the GEMM pass that follows).
// ---------------------------------------------------------------------------
__global__ __launch_bounds__(128)
void rownorm_kernel(const float* __restrict__ src, const float* __restrict__ tgt,
                    float* __restrict__ ws) {
    int row = blockIdx.x;                               // 0..1599
    const float* base = (row < NB * NS)
        ? (src + (size_t)row * NP)
        : (tgt + (size_t)(row - NB * NS) * NP);
    const float4* p4 = (const float4*)base;
    float acc = 0.f;
    for (int i = threadIdx.x; i < NP / 4; i += 128) {
        float4 x = p4[i];
        acc += x.x * x.x + x.y * x.y + x.z * x.z + x.w * x.w;
    }
    __shared__ float red[128];
    red[threadIdx.x] = acc;
    __syncthreads();
    for (int s = 64; s > 0; s >>= 1) {
        if (threadIdx.x < s) red[threadIdx.x] += red[threadIdx.x + s];
        __syncthreads();
    }
    if (threadIdx.x == 0) ws[row] = red[0];
}

// ---------------------------------------------------------------------------
// Kernel 2: cost[b,m,n] = s2[m] + t2[n] - 2 * <src[b,m,:], tgt[b,n,:]>
// One 512-thread block (16 wave32s) per 16x16 output tile. Each wave runs
// V_WMMA_F32_16X16X4_F32 over a disjoint 2304-wide K-slice (4096 waves total
// in flight -> enough MLP to reach HBM bandwidth), then the 16 partial
// accumulators are combined in LDS in a fixed order (deterministic).
// A-frag (16x4 f32): lane%16 = M, VGPR0/1 hold K = 2*(lane/16) + {0,1}.
// B-frag (4x16 f32): lane%16 = N, same K striping (B taken as tgt^T).
// ---------------------------------------------------------------------------
__global__ __launch_bounds__(512)
void cost_wmma_kernel(const float* __restrict__ src, const float* __restrict__ tgt,
                      const float* __restrict__ ws, float* __restrict__ cost) {
    int blk   = blockIdx.x;          // 256 blocks = 16 batches * 4 * 4 tiles
    int b     = blk >> 4;
    int tile  = blk & 15;
    int mtile = tile >> 2;
    int ntile = tile & 3;
    int wave  = threadIdx.x >> 5;    // 0..15 -> K-slice
    int lane  = threadIdx.x & 31;
    int half  = lane >> 4;           // 0 or 1 -> K offset 0 or 2
    int lid   = lane & 15;

    // Clamp row indices so the 50-row matrix fills the 64-row tile grid
    // without divergent loads (EXEC must be all ones around WMMA).
    int mrow = mtile * 16 + lid; if (mrow > NS - 1) mrow = NS - 1;
    int nrow = ntile * 16 + lid; if (nrow > NS - 1) nrow = NS - 1;

    const float* aptr = src + ((size_t)b * NS + mrow) * NP + wave * KSLICE + 2 * half;
    const float* bptr = tgt + ((size_t)b * NS + nrow) * NP + wave * KSLICE + 2 * half;

    v8f c = {};
#pragma unroll 8
    for (int k = 0; k < KSLICE; k += 4) {
        v2f a  = *(const v2f*)(aptr + k);
        v2f bb = *(const v2f*)(bptr + k);
        // 8 args: (neg_a, A, neg_b, B, c_mod, C, reuse_a, reuse_b)
        c = __builtin_amdgcn_wmma_f32_16x16x4_f32(
                false, a, false, bb, (short)0, c, false, false);
    }

    // Deterministic cross-wave combine of the 16 K-slice partials.
    __shared__ float partial[KSPLIT][256];
#pragma unroll
    for (int r = 0; r < 8; ++r) partial[wave][lane * 8 + r] = c[r];
    __syncthreads();

    int t = threadIdx.x;
    if (t < 256) {
        float dot = 0.f;
#pragma unroll
        for (int w = 0; w < KSPLIT; ++w) dot += partial[w][t];

        // C-frag layout: lane%16 = N, VGPR r -> M = r + 8*(lane/16)
        int lane2 = t >> 3;
        int r     = t & 7;
        int m_g   = mtile * 16 + r + 8 * (lane2 >> 4);
        int n_g   = ntile * 16 + (lane2 & 15);
        if (m_g < NS && n_g < NS) {
            const float* s2 = ws + (size_t)b * NS;
            const float* t2 = ws + NB * NS + (size_t)b * NS;
            cost[((size_t)b * NS + m_g) * NS + n_g] =
                s2[m_g] + t2[n_g] - 2.0f * dot;
        }
    }
}

// ---------------------------------------------------------------------------
// Kernel 3: Jonker-Volgenant shortest augmenting path, one wave32 per batch.
// Column scans (minv update + argmin with first-occurrence tie-break, matching
// np.argmin) are lane-parallel in LDS; augmentation is serial on lane 0.
// ---------------------------------------------------------------------------
#define HINF 1e30f

__global__ __launch_bounds__(32)
void hungarian_kernel(const float* __restrict__ cost,
                      float* __restrict__ out_src, float* __restrict__ out_tgt) {
    int b   = blockIdx.x;
    int tid = threadIdx.x;

    __shared__ float c[NS][NS];
    __shared__ float u[NS + 1], v[NS + 1], minv[NS + 1];
    __shared__ int   way[NS + 1], p[NS + 1], usedArr[NS + 1];
    __shared__ float redV[32];
    __shared__ int   redI[32];
    __shared__ float s_delta;
    __shared__ int   s_j0;

    for (int idx = tid; idx < NS * NS; idx += 32)
        c[idx / NS][idx % NS] = cost[(size_t)b * NS * NS + idx];
    for (int j = tid; j <= NS; j += 32) {
        u[j] = 0.f; v[j] = 0.f; p[j] = 0; way[j] = 0;
    }
    __syncthreads();

    for (int i = 1; i <= NS; ++i) {
        for (int j = tid; j <= NS; j += 32) { minv[j] = HINF; usedArr[j] = 0; }
        if (tid == 0) { p[0] = i; s_j0 = 0; }
        __syncthreads();

        while (true) {
            int j0 = s_j0;
            if (tid == 0) usedArr[j0] = 1;
            __syncthreads();

            int   i0   = p[j0];
            float u_i0 = u[i0];

            float bestV = HINF; int bestJ = NS + 1;
            for (int j = tid + 1; j <= NS; j += 32) {
                if (!usedArr[j]) {
                    float cur = c[i0 - 1][j - 1] - u_i0 - v[j];
                    if (cur < minv[j]) { minv[j] = cur; way[j] = j0; }
                    float mv = minv[j];
                    if (mv < bestV || (mv == bestV && j < bestJ)) {
                        bestV = mv; bestJ = j;
                    }
                }
            }
            redV[tid] = bestV; redI[tid] = bestJ;
            __syncthreads();
            if (tid == 0) {
                float dV = HINF; int dJ = NS + 1;
                for (int l = 0; l < 32; ++l) {
                    if (redV[l] < dV || (redV[l] == dV && redI[l] < dJ)) {
                        dV = redV[l]; dJ = redI[l];
                    }
                }
                s_delta = dV; s_j0 = dJ;
            }
            __syncthreads();

            float delta = s_delta;
            int   j1    = s_j0;
            for (int j = tid; j <= NS; j += 32) {
                if (usedArr[j]) { u[p[j]] += delta; v[j] -= delta; }
                else            { minv[j] -= delta; }
            }
            __syncthreads();

            if (p[j1] == 0) break;     // uniform: all lanes read same LDS
        }

        if (tid == 0) {                 // augment along 'way' chain
            int j0 = s_j0;
            while (j0) { int j1 = way[j0]; p[j0] = p[j1]; j0 = j1; }
        }
        __syncthreads();
    }

    for (int j = tid + 1; j <= NS; j += 32) {
        int r = p[j];
        if (r > 0) out_tgt[b * NS + (r - 1)] = (float)(j - 1);
    }
    for (int r = tid; r < NS; r += 32) out_src[b * NS + r] = (float)r;
}

// ---------------------------------------------------------------------------
extern "C" void kernel_launch(void* const* d_in, const int* in_sizes, int n_in,
                              void* d_out, int out_size, void* d_ws, size_t ws_size,
                              hipStream_t stream) {
    const float* src = (const float*)d_in[0];   // [16,50,192,192] f32
    const float* tgt = (const float*)d_in[1];   // [16,50,192,192] f32
    float* out = (float*)d_out;                 // cost(40000) | src_idx(800) | tgt_idx(800)
    float* ws  = (float*)d_ws;                  // 1600 floats: s2 | t2

    rownorm_kernel<<<2 * NB * NS, 128, 0, stream>>>(src, tgt, ws);
    cost_wmma_kernel<<<NB * 16, 512, 0, stream>>>(src, tgt, ws, out);
    hungarian_kernel<<<NB, 32, 0, stream>>>(out, out + NB * NS * NS,
                                            out + NB * NS * NS + NB * NS);
}